// AttentionPredictor_33449205301963
// MI455X (gfx1250) — compile-verified
//
#include <hip/hip_runtime.h>

typedef __attribute__((ext_vector_type(2))) float v2f;
typedef __attribute__((ext_vector_type(8))) float v8f;

// ---------------------------------------------------------------------------
// Phase 1: rowsum[n] = sum_f h[n, f]  via V_WMMA_F32_16X16X4_F32 with B = ones
//   - block = 256 threads (8 waves, wave32), 128 rows per block
//   - 64 KB LDS tile, XOR-swizzled so half-wave A-fragment reads are
//     bank-conflict-free (banks = col ^ 4*(row&15), 16 distinct)
// ---------------------------------------------------------------------------
__global__ __launch_bounds__(256) void rowsum_wmma_kernel(
    const float* __restrict__ h, float* __restrict__ rowsum, int N) {
  __shared__ float tile[128 * 128];  // exactly 64 KB

  const int tid  = threadIdx.x;
  const int base = blockIdx.x * 128;

  // ---- Stage 128 rows x 128 cols into LDS, coalesced float4 ----
  // 128 rows * 32 float4/row = 4096 float4; 256 threads -> 16 each.
#pragma unroll
  for (int i = 0; i < 16; ++i) {
    const int idx  = tid + i * 256;  // float4 index within tile
    const int row  = idx >> 5;       // 32 float4 per row
    const int col4 = idx & 31;
    int grow = base + row;
    if (grow >= N) grow = N - 1;     // clamp; clamped results never stored
    const float4 v =
        reinterpret_cast<const float4*>(h + (size_t)grow * 128)[col4];
    const int scol = (col4 * 4) ^ ((row & 15) << 2);  // XOR bank swizzle
    *reinterpret_cast<float4*>(&tile[row * 128 + scol]) = v;
  }
  __syncthreads();

  // ---- Each wave reduces its 16 rows with 32 chained WMMAs ----
  const int lane = tid & 31;
  const int wave = tid >> 5;                 // 0..7
  const int rl   = lane & 15;                // row within wave tile
  const int r    = (wave << 4) + rl;         // local row 0..127
  const int koff = (lane >> 4) << 1;         // 0 (lanes 0-15) or 2 (16-31)
  const int sw   = rl << 2;                  // per-row XOR swizzle

  v2f ones;
  ones.x = 1.0f;
  ones.y = 1.0f;
  v8f c = {};
  const float* rp = &tile[r * 128];
#pragma unroll
  for (int k = 0; k < 128; k += 4) {
    const int cc = (k + koff) ^ sw;  // XOR preserves contiguity of the b64 pair
    v2f a;
    a.x = rp[cc];
    a.y = rp[cc + 1];
    // D = A(16x4) x ones(4x16) + C : every column of D holds the row-sum
    c = __builtin_amdgcn_wmma_f32_16x16x4_f32(false, a, false, ones, (short)0,
                                              c, false, false);
  }

  // C/D layout: lanes 0-15 vgpr i -> row i; lanes 16-31 vgpr i -> row 8+i.
  // Column value is identical for all N, so lane 0 / lane 16 store 8 rows each.
  if (rl == 0) {
    const int half  = lane >> 4;  // 0 or 1
    const int grow0 = base + (wave << 4) + (half << 3);
    const float vals[8] = {c[0], c[1], c[2], c[3], c[4], c[5], c[6], c[7]};
#pragma unroll
    for (int i = 0; i < 8; ++i) {
      const int g = grow0 + i;
      if (g < N) rowsum[g] = vals[i];
    }
  }
}

// ---------------------------------------------------------------------------
// Phase 2: out[e] = rowsum[edge_src[e]]   (gate == softmax over size-1 == 1)
// Vectorized x4; rowsum table (400 KB) is L2-resident.
// ---------------------------------------------------------------------------
__global__ __launch_bounds__(256) void gather_rowsum_kernel(
    const float* __restrict__ rowsum, const int* __restrict__ edge_src,
    float* __restrict__ out, int E) {
  int i = (blockIdx.x * 256 + threadIdx.x) * 4;
  if (i + 3 < E) {
    const int4 s = *reinterpret_cast<const int4*>(edge_src + i);
    float4 o;
    o.x = rowsum[s.x];
    o.y = rowsum[s.y];
    o.z = rowsum[s.z];
    o.w = rowsum[s.w];
    *reinterpret_cast<float4*>(out + i) = o;
  } else {
    for (; i < E; ++i) out[i] = rowsum[edge_src[i]];
  }
}

extern "C" void kernel_launch(void* const* d_in, const int* in_sizes, int n_in,
                              void* d_out, int out_size, void* d_ws,
                              size_t ws_size, hipStream_t stream) {
  // inputs: h [N*128] f32, W [256] f32 (dead), b [1] f32 (dead),
  //         edge_src [E] i32, edge_dst [E] i32 (dead)
  const float* h        = (const float*)d_in[0];
  const int*   edge_src = (const int*)d_in[3];
  const int    N        = in_sizes[0] / 128;
  const int    E        = in_sizes[3];

  float* rowsum = (float*)d_ws;  // N*4 bytes of scratch

  const int grid1 = (N + 127) / 128;
  rowsum_wmma_kernel<<<grid1, 256, 0, stream>>>(h, rowsum, N);

  const int t4    = (E + 3) / 4;
  const int grid2 = (t4 + 255) / 256;
  gather_rowsum_kernel<<<grid2, 256, 0, stream>>>(rowsum, edge_src,
                                                  (float*)d_out, E);
}